// FBTTEmbedding_72825465471568
// MI455X (gfx1250) — compile-verified
//
#include <hip/hip_runtime.h>

// P = (100,100,100), Q = (4,4,8), R = (1,32,32,1), N = 1<<20
#define P0 100
#define P1 100
#define P2 100

typedef __attribute__((ext_vector_type(2))) float v2f;
typedef __attribute__((ext_vector_type(8))) float v8f;

// ---------------------------------------------------------------------------
// Phase 1: build ab[i0][i1][16][32]  (20.48 MB, L2-resident) in d_ws.
//   For each i1: T[400,128] = A[400,32] @ core1[i1][32,128]
//   where A row (i0*4+q0) = core0[i0] reshaped [4,32].
//   Scatter T into ab layout: ab[i0][i1][q0*4+q1][r2] = T[i0*4+q0][q1*32+r2].
// One wave per (i1, mtile): 16 rows of A x 128 cols, K=32 -> 8x8 = 64 WMMA.
// ---------------------------------------------------------------------------
__global__ __launch_bounds__(256) void tt_phase1(const float* __restrict__ core0,
                                                 const float* __restrict__ core1,
                                                 float* __restrict__ ab) {
  const int lane = threadIdx.x & 31;
  const int wave = blockIdx.x * (blockDim.x >> 5) + (threadIdx.x >> 5);
  if (wave >= P1 * 25) return;                 // wave-uniform: EXEC stays all-1s
  const int i1    = wave / 25;
  const int mtile = wave % 25;

  const int mrow  = lane & 15;                 // A-layout: lane -> M row
  const int khalf = lane >> 4;                 // lanes 16-31 hold K+2
  const int grow  = mtile * 16 + mrow;         // global A row 0..399

  // A operand: 8 K-subtiles, float2 (consecutive K) per lane, 8B aligned
  v2f aop[8];
  const float* arow = core0 + grow * 32;       // A row r = core0 flat + r*32
#pragma unroll
  for (int t = 0; t < 8; ++t)
    aop[t] = *(const v2f*)(arow + 4 * t + 2 * khalf);

  const float* b = core1 + (size_t)i1 * 4096;  // [32][128] row-major

  for (int nt = 0; nt < 8; ++nt) {             // 8 N-tiles of 16 cols
    const int ncol = nt * 16 + (lane & 15);    // B-layout: lane -> N col
    v8f acc = {};
#pragma unroll
    for (int t = 0; t < 8; ++t) {
      const int k0 = 4 * t + 2 * khalf;        // consecutive K per lane
      v2f bop;
      bop.x = b[k0 * 128 + ncol];
      bop.y = b[(k0 + 1) * 128 + ncol];
      acc = __builtin_amdgcn_wmma_f32_16x16x4_f32(false, aop[t], false, bop,
                                                  (short)0, acc, false, false);
    }
    // D layout: VGPR j -> tile row (j + 8*khalf), col = ncol. Scatter to ab.
#pragma unroll
    for (int j = 0; j < 8; ++j) {
      const int M  = mtile * 16 + j + 8 * khalf;   // = i0*4 + q0
      const int i0 = M >> 2;
      const int q0 = M & 3;
      const int q1 = ncol >> 5;
      const int r2 = ncol & 31;
      ab[((size_t)i0 * P1 + i1) * 512 + (q0 * 4 + q1) * 32 + r2] = acc[j];
    }
  }
}

// ---------------------------------------------------------------------------
// Phase 2: one wave per index.
//   out[n][16,8] = ab[i0,i1][16,32] @ core2[i2][32,8]
// 8x v_wmma_f32_16x16x4_f32 accumulating over K=32. N padded 8->16 by
// duplicating c columns (cols 8-15 of D discarded at store) — branchless,
// EXEC all-1s at every WMMA. Output written non-temporally (512 MB stream,
// keep the 20 MB ab table hot in L2).
// ---------------------------------------------------------------------------
__global__ __launch_bounds__(256) void tt_phase2(const float* __restrict__ ab,
                                                 const float* __restrict__ core2,
                                                 const int* __restrict__ indices,
                                                 float* __restrict__ out,
                                                 int n_total) {
  const int lane = threadIdx.x & 31;
  const int wave = blockIdx.x * (blockDim.x >> 5) + (threadIdx.x >> 5);
  if (wave >= n_total) return;                 // wave-uniform

  const int idx  = indices[wave];              // 0..999999
  const int i2   = idx % P2;
  const int rest = idx / P2;
  const int i1   = rest % P1;
  const int i0   = rest / P1;

  const float* tp = ab + ((size_t)i0 * P1 + i1) * 512;  // [16][32]
  const float* c  = core2 + (size_t)i2 * 256;           // [32][8]

  const int mrow  = lane & 15;
  const int khalf = lane >> 4;
  const int ncol  = lane & 15;
  const int ccol  = ncol & 7;                  // duplicate c cols for N pad

  v8f acc = {};
#pragma unroll
  for (int t = 0; t < 8; ++t) {
    const int k0 = 4 * t + 2 * khalf;
    v2f aop = *(const v2f*)(tp + mrow * 32 + k0);       // aligned b64
    v2f bop;
    bop.x = c[k0 * 8 + ccol];
    bop.y = c[(k0 + 1) * 8 + ccol];
    acc = __builtin_amdgcn_wmma_f32_16x16x4_f32(false, aop, false, bop,
                                                (short)0, acc, false, false);
  }

  float* o = out + (size_t)wave * 128;         // out[n][row*8 + col]
  if (ncol < 8) {                              // divergence only after WMMAs
#pragma unroll
    for (int j = 0; j < 8; ++j) {
      const int row = j + 8 * khalf;
      __builtin_nontemporal_store(acc[j], o + row * 8 + ncol);
    }
  }
}

// ---------------------------------------------------------------------------
extern "C" void kernel_launch(void* const* d_in, const int* in_sizes, int n_in,
                              void* d_out, int out_size, void* d_ws, size_t ws_size,
                              hipStream_t stream) {
  const float* core0 = (const float*)d_in[0];   // [100,128]
  const float* core1 = (const float*)d_in[1];   // [100,4096]
  const float* core2 = (const float*)d_in[2];   // [100,256]
  const int* indices = (const int*)d_in[3];     // [N]
  float* out = (float*)d_out;                   // [N,128]
  float* ab  = (float*)d_ws;                    // 100*100*512 floats = 20.48 MB
  const int n = in_sizes[3];

  // Phase 1: 2500 waves (100 i1 x 25 M-tiles), 8 waves/block
  tt_phase1<<<dim3((P1 * 25 + 7) / 8), dim3(256), 0, stream>>>(core0, core1, ab);

  // Phase 2: one wave per index
  tt_phase2<<<dim3((n + 7) / 8), dim3(256), 0, stream>>>(ab, core2, indices, out, n);
}